// GCNDiff_53712861003990
// MI455X (gfx1250) — compile-verified
//
#include <hip/hip_runtime.h>
#include <hip/hip_bf16.h>

typedef __attribute__((ext_vector_type(2))) float v2f;
typedef __attribute__((ext_vector_type(8))) float v8f;

#define F_IN  256
#define HID   16
#define C_OUT 40

// ---------------------------------------------------------------------------
// GEMM1: h1[N,16] = x[N,256] @ W1[256,16]   (f32 WMMA 16x16x4, K-loop of 64)
// One wave32 per 16x16 output tile; W1 staged in LDS (16 KB) per block.
// ---------------------------------------------------------------------------
__global__ __launch_bounds__(256) void gemm1_wmma(
    const float* __restrict__ x, const float* __restrict__ W1,
    float* __restrict__ h1, int n_mtiles) {
  __shared__ float sW[F_IN * HID];  // 16 KB
  const int t = threadIdx.x;
  for (int i = t; i < F_IN * HID; i += 256) sW[i] = W1[i];
  __syncthreads();

  const int wave = t >> 5;
  const int lane = t & 31;
  const int tile = blockIdx.x * 8 + wave;
  if (tile >= n_mtiles) return;            // whole-wave uniform: EXEC stays all-1s

  const int mbase = tile * 16;
  const int mrow  = mbase + (lane & 15);
  const int khalf = (lane >> 4) * 2;       // A layout: lanes 0-15 -> K=k,k+1 ; 16-31 -> K=k+2,k+3
  const int ncol  = lane & 15;
  const float* xrow = x + (size_t)mrow * F_IN + khalf;

  v8f c = {};
  for (int k = 0; k < F_IN; k += 4) {
    v2f a = *(const v2f*)(xrow + k);       // 8B-aligned (khalf even, k%4==0)
    v2f b;
    b.x = sW[(k + khalf) * HID + ncol];
    b.y = sW[(k + khalf + 1) * HID + ncol];
    c = __builtin_amdgcn_wmma_f32_16x16x4_f32(false, a, false, b, (short)0, c,
                                              false, false);
  }
  // C/D layout: VGPR v -> row mbase+v (lanes 0-15) / mbase+v+8 (lanes 16-31)
  const int moff = (lane >> 4) * 8;
#pragma unroll
  for (int v = 0; v < 8; ++v)
    h1[(size_t)(mbase + v + moff) * HID + ncol] = c[v];
}

// ---------------------------------------------------------------------------
// GEMM2: h2[N,40] = h1b[N,16] @ W2[16,40]   (N padded to 3x16, K=16 -> 4 WMMA)
// ---------------------------------------------------------------------------
__global__ __launch_bounds__(256) void gemm2_wmma(
    const float* __restrict__ h1b, const float* __restrict__ W2,
    float* __restrict__ h2, int n_mtiles) {
  __shared__ float sW[HID * 48];  // W2 zero-padded to 16x48
  const int t = threadIdx.x;
  for (int i = t; i < HID * 48; i += 256) {
    int k = i / 48, n = i % 48;
    sW[i] = (n < C_OUT) ? W2[k * C_OUT + n] : 0.0f;
  }
  __syncthreads();

  const int wave  = t >> 5;
  const int lane  = t & 31;
  const int gtile = blockIdx.x * 8 + wave;
  if (gtile >= n_mtiles * 3) return;

  const int mtile = gtile / 3, ntile = gtile % 3;
  const int mbase = mtile * 16, nbase = ntile * 16;
  const int mrow  = mbase + (lane & 15);
  const int khalf = (lane >> 4) * 2;
  const int nloc  = lane & 15;
  const float* arow = h1b + (size_t)mrow * HID + khalf;

  v8f c = {};
#pragma unroll
  for (int k = 0; k < HID; k += 4) {
    v2f a = *(const v2f*)(arow + k);
    v2f b;
    b.x = sW[(k + khalf) * 48 + nbase + nloc];
    b.y = sW[(k + khalf + 1) * 48 + nbase + nloc];
    c = __builtin_amdgcn_wmma_f32_16x16x4_f32(false, a, false, b, (short)0, c,
                                              false, false);
  }
  const int moff = (lane >> 4) * 8;
  const int ncol = nbase + nloc;
  if (ncol < C_OUT) {
#pragma unroll
    for (int v = 0; v < 8; ++v)
      h2[(size_t)(mbase + v + moff) * C_OUT + ncol] = c[v];
  }
}

// ---------------------------------------------------------------------------
// Zero-fill (agg buffers must be cleared every call; harness poisons ws)
// ---------------------------------------------------------------------------
__global__ void zero_f32(float* __restrict__ p, long long n) {
  long long i = (long long)blockIdx.x * blockDim.x + threadIdx.x;
  if (i < n) p[i] = 0.0f;
}

// ---------------------------------------------------------------------------
// Scatter layer 1: 4 threads/edge, float4 gather from L2-resident h1,
// 4x global_atomic_add_f32 into agg1[dst].
// ---------------------------------------------------------------------------
__global__ void scatter1(const long long* __restrict__ ei,
                         const float* __restrict__ ew,
                         const float* __restrict__ h1,
                         float* __restrict__ agg1, long long E) {
  long long idx = (long long)blockIdx.x * blockDim.x + threadIdx.x;
  if (idx >= E * 4) return;
  long long e = idx >> 2;
  int sub = (int)(idx & 3);
  long long s = ei[e];
  long long d = ei[E + e];
  float w = ew[e];
  float4 m = *(const float4*)(h1 + s * HID + sub * 4);
  float* o = agg1 + d * HID + sub * 4;
  atomicAdd(o + 0, m.x * w);
  atomicAdd(o + 1, m.y * w);
  atomicAdd(o + 2, m.z * w);
  atomicAdd(o + 3, m.w * w);
}

// ---------------------------------------------------------------------------
// Scatter layer 2: 10 threads/edge (40 floats)
// ---------------------------------------------------------------------------
__global__ void scatter2(const long long* __restrict__ ei,
                         const float* __restrict__ ew,
                         const float* __restrict__ h2,
                         float* __restrict__ agg2, long long E) {
  long long idx = (long long)blockIdx.x * blockDim.x + threadIdx.x;
  if (idx >= E * 10) return;
  long long e = idx / 10;
  int sub = (int)(idx - e * 10);
  long long s = ei[e];
  long long d = ei[E + e];
  float w = ew[e];
  float4 m = *(const float4*)(h2 + s * C_OUT + sub * 4);
  float* o = agg2 + d * C_OUT + sub * 4;
  atomicAdd(o + 0, m.x * w);
  atomicAdd(o + 1, m.y * w);
  atomicAdd(o + 2, m.z * w);
  atomicAdd(o + 3, m.w * w);
}

// ---------------------------------------------------------------------------
// bias + ReLU (h1b written back over h1 buffer)
// ---------------------------------------------------------------------------
__global__ void bias_relu(const float* __restrict__ agg1,
                          const float* __restrict__ b1,
                          float* __restrict__ h1b, long long n) {
  long long i = (long long)blockIdx.x * blockDim.x + threadIdx.x;
  if (i < n) {
    float v = agg1[i] + b1[(int)(i & (HID - 1))];
    h1b[i] = fmaxf(v, 0.0f);
  }
}

// ---------------------------------------------------------------------------
// bias + log_softmax over C=40 (fully unrolled, row in registers)
// ---------------------------------------------------------------------------
__global__ void bias_logsoftmax(const float* __restrict__ agg2,
                                const float* __restrict__ b2,
                                float* __restrict__ out, int n_nodes) {
  int node = blockIdx.x * blockDim.x + threadIdx.x;
  if (node >= n_nodes) return;
  const float* row = agg2 + (size_t)node * C_OUT;
  float z[C_OUT];
  float mx = -__builtin_inff();
#pragma unroll
  for (int c = 0; c < C_OUT; ++c) {
    z[c] = row[c] + b2[c];
    mx = fmaxf(mx, z[c]);
  }
  float s = 0.0f;
#pragma unroll
  for (int c = 0; c < C_OUT; ++c) s += __expf(z[c] - mx);
  float lse = mx + __logf(s);
  float* orow = out + (size_t)node * C_OUT;
#pragma unroll
  for (int c = 0; c < C_OUT; ++c) orow[c] = z[c] - lse;
}

extern "C" void kernel_launch(void* const* d_in, const int* in_sizes, int n_in,
                              void* d_out, int out_size, void* d_ws, size_t ws_size,
                              hipStream_t stream) {
  const float*     x   = (const float*)d_in[0];
  const long long* ei  = (const long long*)d_in[1];   // int64 edge_index [2,E]
  const float*     ew  = (const float*)d_in[2];
  const float*     W1  = (const float*)d_in[3];
  const float*     b1  = (const float*)d_in[4];
  const float*     W2  = (const float*)d_in[5];
  const float*     b2  = (const float*)d_in[6];
  float*           out = (float*)d_out;

  const int       n = in_sizes[0] / F_IN;       // 100000 (multiple of 16)
  const long long E = (long long)in_sizes[1] / 2;  // 3.2M

  // Workspace: h1 | agg1 | h2 | agg2  (~45 MB)
  float* h1   = (float*)d_ws;
  float* agg1 = h1 + (size_t)n * HID;
  float* h2   = agg1 + (size_t)n * HID;
  float* agg2 = h2 + (size_t)n * C_OUT;

  const int n_mtiles = (n + 15) / 16;  // 6250

  // Layer 1
  gemm1_wmma<<<(n_mtiles + 7) / 8, 256, 0, stream>>>(x, W1, h1, n_mtiles);
  {
    long long cnt = (long long)n * HID;
    zero_f32<<<(unsigned)((cnt + 255) / 256), 256, 0, stream>>>(agg1, cnt);
  }
  {
    long long tot = E * 4;
    scatter1<<<(unsigned)((tot + 255) / 256), 256, 0, stream>>>(ei, ew, h1, agg1, E);
  }
  {
    long long cnt = (long long)n * HID;
    bias_relu<<<(unsigned)((cnt + 255) / 256), 256, 0, stream>>>(agg1, b1, h1, cnt);
  }

  // Layer 2
  gemm2_wmma<<<(n_mtiles * 3 + 7) / 8, 256, 0, stream>>>(h1, W2, h2, n_mtiles);
  {
    long long cnt = (long long)n * C_OUT;
    zero_f32<<<(unsigned)((cnt + 255) / 256), 256, 0, stream>>>(agg2, cnt);
  }
  {
    long long tot = E * 10;
    scatter2<<<(unsigned)((tot + 255) / 256), 256, 0, stream>>>(ei, ew, h2, agg2, E);
  }
  bias_logsoftmax<<<(n + 255) / 256, 256, 0, stream>>>(agg2, b2, out, n);
}